// BorderBranch_11982958756178
// MI455X (gfx1250) — compile-verified
//
#include <hip/hip_runtime.h>
#include <hip/hip_bf16.h>

// ---------------------------------------------------------------------------
// BorderBranch on MI455X (gfx1250, wave32)
//   conv1x1 -> InstanceNorm -> ReLU   (x2, GEMM via V_WMMA_F32_16X16X4_F32)
//   border_align (bilinear + max over 11 border samples, channel-minor gathers)
//   conv1x1 on concat -> ReLU         (WMMA, K=640)
// Roofline: ~6 GFLOP fp32, ~60 MB working set -> fully L2-resident (192 MB).
// Bound by L2 traffic, so: 32x64 WMMA strips (2x B reuse vs 16x64) and a
// channel-minor ltrb copy so border gathers are lane-contiguous.
// ---------------------------------------------------------------------------

typedef __attribute__((ext_vector_type(2))) float v2f;
typedef __attribute__((ext_vector_type(8))) float v8f;

#define NBATCH 2
#define CIN    256
#define HDIM   64
#define WDIM   64
#define HW     4096          // 64*64
#define CCUR   128           // conv_cur out channels
#define CLTRB  512           // conv_ltrb out channels (4 * 128)
#define CTOT   640           // CCUR + CLTRB rows in Y buffer
#define KCAT   640           // concat channels into final conv
#define MOUT   256           // final conv out channels
#define POOLP  11            // pool_size + 1 samples per border

// ---------------------------------------------------------------------------
// Kernel 1: Y[n][obase+o][s] = sum_c Wm[o][c] * X[n][c][s] + bias[o]
// One wave computes a 32(M) x 64(N) strip: 2 A fragments x 4 B tiles share
// loads across 8 accumulators. fp32 WMMA: A 16x4, B 4x16, C/D 16x16.
// ---------------------------------------------------------------------------
__global__ __launch_bounds__(32)
void conv1x1_wmma_kernel(const float* __restrict__ X,   // [n][CIN][HW]
                         const float* __restrict__ Wm,  // [M][CIN]
                         const float* __restrict__ bias,// [M]
                         float* __restrict__ Y,         // [n][CTOT][HW]
                         int obase) {
    const int lane = threadIdx.x;
    const int half = lane >> 4;      // 0: K pair {0,1}, 1: K pair {2,3}
    const int l16  = lane & 15;
    const int s0   = blockIdx.x * 64;
    const int m0   = blockIdx.y * 32;
    const int n    = blockIdx.z;

    const float* Xn = X + (size_t)n * CIN * HW;

    v8f acc[2][4];
#pragma unroll
    for (int u = 0; u < 2; ++u) {
#pragma unroll
        for (int r = 0; r < 8; ++r) {
            // C/D layout: VGPR r holds M = r (lanes 0-15) / r+8 (lanes 16-31)
            float bv = bias[m0 + u * 16 + r + half * 8];
            acc[u][0][r] = bv; acc[u][1][r] = bv;
            acc[u][2][r] = bv; acc[u][3][r] = bv;
        }
    }

    // A fragments: rows m0+l16 and m0+16+l16; lane covers K = k + half*2 + {0,1}
    const float* wp0 = Wm + (size_t)(m0 + l16) * CIN + half * 2;
    const float* wp1 = wp0 + (size_t)16 * CIN;

    for (int k = 0; k < CIN; k += 4) {
        v2f a0, a1;
        a0.x = wp0[k]; a0.y = wp0[k + 1];
        a1.x = wp1[k]; a1.y = wp1[k + 1];
        const float* xp = Xn + (size_t)(k + half * 2) * HW + s0 + l16;
#pragma unroll
        for (int t = 0; t < 4; ++t) {
            v2f b;
            b.x = xp[t * 16];        // row k+half*2,   col s0+t*16+l16
            b.y = xp[t * 16 + HW];   // row k+half*2+1
            acc[0][t] = __builtin_amdgcn_wmma_f32_16x16x4_f32(
                false, a0, false, b, (short)0, acc[0][t], false, false);
            acc[1][t] = __builtin_amdgcn_wmma_f32_16x16x4_f32(
                false, a1, false, b, (short)0, acc[1][t], false, false);
        }
    }

#pragma unroll
    for (int u = 0; u < 2; ++u) {
        float* yp = Y + ((size_t)n * CTOT + obase + m0 + u * 16) * HW + s0 + l16;
#pragma unroll
        for (int r = 0; r < 8; ++r) {
            size_t row = (size_t)(r + half * 8) * HW;
#pragma unroll
            for (int t = 0; t < 4; ++t) yp[row + t * 16] = acc[u][t][r];
        }
    }
}

// ---------------------------------------------------------------------------
// Kernel 2: InstanceNorm (biased var, eps=1e-5) + ReLU.
// One 256-thread block per (n, channel) row of 4096 elements.
//  - cur rows (ch <  CCUR): normalized in place (consumed by final conv)
//  - ltrb rows (ch >= CCUR): written ONLY to the channel-minor copy
//    LT[n][hw][512] (consumed only by border_align; lane-contiguous gathers)
// ---------------------------------------------------------------------------
__global__ __launch_bounds__(256)
void inorm_relu_kernel(float* __restrict__ Y,   // [n][CTOT][HW]
                       float* __restrict__ LT) {// [n][HW][CLTRB]
    const int row = blockIdx.x;          // 0 .. NBATCH*CTOT-1
    const int n   = row / CTOT;
    const int ch  = row % CTOT;
    float* p = Y + (size_t)row * HW;
    const int tid = threadIdx.x;

    float s = 0.f, s2 = 0.f;
    for (int i = tid; i < HW; i += 256) {
        float v = p[i];
        s  += v;
        s2 += v * v;
    }
    // wave32 reduction
#pragma unroll
    for (int off = 16; off > 0; off >>= 1) {
        s  += __shfl_xor(s,  off, 32);
        s2 += __shfl_xor(s2, off, 32);
    }
    __shared__ float ls[8], ls2[8];
    const int wave = tid >> 5;
    if ((tid & 31) == 0) { ls[wave] = s; ls2[wave] = s2; }
    __syncthreads();
    float S = 0.f, S2 = 0.f;
#pragma unroll
    for (int w = 0; w < 8; ++w) { S += ls[w]; S2 += ls2[w]; }

    const float mean = S * (1.f / HW);
    const float var  = fmaf(-mean, mean, S2 * (1.f / HW));
    const float inv  = rsqrtf(var + 1e-5f);

    if (ch < CCUR) {
        for (int i = tid; i < HW; i += 256)
            p[i] = fmaxf((p[i] - mean) * inv, 0.f);
    } else {
        const int c = ch - CCUR;         // ltrb channel in [0,512)
        float* lt = LT + (size_t)n * HW * CLTRB + c;
        for (int i = tid; i < HW; i += 256)
            lt[(size_t)i * CLTRB] = fmaxf((p[i] - mean) * inv, 0.f);
    }
}

// ---------------------------------------------------------------------------
// Kernel 3: BorderAlign (mmcv semantics). Block = (n, box k, border g),
// thread = channel c in [0,128). Reads channel-minor LT so each bilinear
// corner gather is 128 contiguous floats. Output written directly in the
// torch permute+reshape flat layout: BA[n] flat = cb*16384 + k*4 + border.
// ---------------------------------------------------------------------------
__global__ __launch_bounds__(128)
void border_align_kernel(const float* __restrict__ LT,    // [n][HW][CLTRB]
                         const float* __restrict__ boxes, // [n][HW][4]
                         float* __restrict__ BA) {        // [n][CLTRB][HW] reshaped
    const int k      = blockIdx.x;   // box index (== spatial index, K = HW)
    const int border = blockIdx.y;   // 0=top,1=left,2=bottom,3=right
    const int n      = blockIdx.z;
    const int c      = threadIdx.x;  // channel within border group

    const float* box = boxes + ((size_t)n * HW + k) * 4;
    const float x1 = box[0], y1 = box[1], x2 = box[2], y2 = box[3];
    const float bw = x2 - x1, bh = y2 - y1;

    float sx, sy, dx, dy;
    if (border == 0)      { sx = x1; sy = y1; dx = bw;  dy = 0.f; }  // top
    else if (border == 1) { sx = x1; sy = y1; dx = 0.f; dy = bh;  }  // left
    else if (border == 2) { sx = x1; sy = y2; dx = bw;  dy = 0.f; }  // bottom
    else                  { sx = x2; sy = y1; dx = 0.f; dy = bh;  }  // right

    // ltrb channel for this (border, c); LT is channel-minor
    const float* feat = LT + (size_t)n * HW * CLTRB + (border * CCUR + c);

    float m = -__builtin_inff();
#pragma unroll
    for (int p = 0; p < POOLP; ++p) {
        const float t = (float)p * (1.f / 10.f);
        const float x = sx + dx * t;
        const float y = sy + dy * t;
        const bool valid = (x >= -1.f) && (x <= (float)WDIM) &&
                           (y >= -1.f) && (y <= (float)HDIM);
        const float xc = fmaxf(x, 0.f);
        const float yc = fmaxf(y, 0.f);
        int x0 = min((int)floorf(xc), WDIM - 1);
        int y0 = min((int)floorf(yc), HDIM - 1);
        const int x1i = min(x0 + 1, WDIM - 1);
        const int y1i = min(y0 + 1, HDIM - 1);
        const float lx = (x0 >= WDIM - 1) ? 0.f : (xc - (float)x0);
        const float ly = (y0 >= HDIM - 1) ? 0.f : (yc - (float)y0);
        const float hx = 1.f - lx, hy = 1.f - ly;

        float v = feat[(size_t)(y0  * WDIM + x0 ) * CLTRB] * (hy * hx)
                + feat[(size_t)(y0  * WDIM + x1i) * CLTRB] * (hy * lx)
                + feat[(size_t)(y1i * WDIM + x0 ) * CLTRB] * (ly * hx)
                + feat[(size_t)(y1i * WDIM + x1i) * CLTRB] * (ly * lx);
        v = valid ? v : 0.f;
        m = fmaxf(m, v);
    }

    // torch ba.transpose(0,3,1,2).reshape(N,-1,H,W) flat layout per batch:
    //   flat = cb*(HW*4) + k*4 + border
    BA[(size_t)n * CLTRB * HW + (size_t)c * (HW * 4) + (size_t)k * 4 + border] = m;
}

// ---------------------------------------------------------------------------
// Kernel 4: final conv1x1 over the 640-channel concat + ReLU -> d_out.
// cat channels [0,512) come from BA buffer, [512,640) from fm_short (Y ch 0-127).
// 32x64 strip per wave; K-step 4 never straddles the 512 boundary.
// ---------------------------------------------------------------------------
__global__ __launch_bounds__(32)
void border_out_wmma_kernel(const float* __restrict__ BA,  // [n][CLTRB][HW]
                            const float* __restrict__ Y,   // [n][CTOT][HW]
                            const float* __restrict__ Wm,  // [MOUT][KCAT]
                            const float* __restrict__ bias,// [MOUT]
                            float* __restrict__ out) {     // [n][MOUT][HW]
    const int lane = threadIdx.x;
    const int half = lane >> 4;
    const int l16  = lane & 15;
    const int s0   = blockIdx.x * 64;
    const int m0   = blockIdx.y * 32;
    const int n    = blockIdx.z;

    const float* BAn  = BA + (size_t)n * CLTRB * HW;
    const float* Ycur = Y  + (size_t)n * CTOT  * HW;   // channels 0..127 = fm_short

    v8f acc[2][4];
#pragma unroll
    for (int u = 0; u < 2; ++u) {
#pragma unroll
        for (int r = 0; r < 8; ++r) {
            float bv = bias[m0 + u * 16 + r + half * 8];
            acc[u][0][r] = bv; acc[u][1][r] = bv;
            acc[u][2][r] = bv; acc[u][3][r] = bv;
        }
    }

    const float* wp0 = Wm + (size_t)(m0 + l16) * KCAT + half * 2;
    const float* wp1 = wp0 + (size_t)16 * KCAT;

    for (int k = 0; k < KCAT; k += 4) {
        v2f a0, a1;
        a0.x = wp0[k]; a0.y = wp0[k + 1];
        a1.x = wp1[k]; a1.y = wp1[k + 1];
        const int kr = k + half * 2;           // kr, kr+1 are this lane's B rows
        const float* bp = (kr < CLTRB) ? (BAn + (size_t)kr * HW)
                                       : (Ycur + (size_t)(kr - CLTRB) * HW);
#pragma unroll
        for (int t = 0; t < 4; ++t) {
            v2f b;
            b.x = bp[s0 + t * 16 + l16];
            b.y = bp[s0 + t * 16 + l16 + HW];
            acc[0][t] = __builtin_amdgcn_wmma_f32_16x16x4_f32(
                false, a0, false, b, (short)0, acc[0][t], false, false);
            acc[1][t] = __builtin_amdgcn_wmma_f32_16x16x4_f32(
                false, a1, false, b, (short)0, acc[1][t], false, false);
        }
    }

#pragma unroll
    for (int u = 0; u < 2; ++u) {
        float* op = out + ((size_t)n * MOUT + m0 + u * 16) * HW + s0 + l16;
#pragma unroll
        for (int r = 0; r < 8; ++r) {
            size_t row = (size_t)(r + half * 8) * HW;
#pragma unroll
            for (int t = 0; t < 4; ++t)
                op[row + t * 16] = fmaxf(acc[u][t][r], 0.f);
        }
    }
}

// ---------------------------------------------------------------------------
extern "C" void kernel_launch(void* const* d_in, const int* in_sizes, int n_in,
                              void* d_out, int out_size, void* d_ws, size_t ws_size,
                              hipStream_t stream) {
    const float* feature  = (const float*)d_in[0];  // [2,256,64,64]
    const float* boxes    = (const float*)d_in[1];  // [2,4096,4]
    const float* w_cur    = (const float*)d_in[2];  // [128,256]
    const float* b_cur    = (const float*)d_in[3];  // [128]
    const float* w_ltrb   = (const float*)d_in[4];  // [512,256]
    const float* b_ltrb   = (const float*)d_in[5];  // [512]
    const float* w_border = (const float*)d_in[6];  // [256,640]
    const float* b_border = (const float*)d_in[7];  // [256]
    float* out = (float*)d_out;                     // [2,256,64,64]

    // Workspace layout (floats):
    //   Y  [2][640][4096]  pre-norm conv outputs (cur normalized in place)
    //   BA [2][512][4096]  border-align result, torch-reshaped layout
    //   LT [2][4096][512]  channel-minor normalized ltrb (gather-friendly)
    float* Y  = (float*)d_ws;
    float* BA = Y  + (size_t)NBATCH * CTOT  * HW;
    float* LT = BA + (size_t)NBATCH * CLTRB * HW;

    const dim3 blk32(32, 1, 1);

    // conv_cur  -> Y channels [0,128)
    conv1x1_wmma_kernel<<<dim3(HW / 64, CCUR / 32, NBATCH), blk32, 0, stream>>>(
        feature, w_cur, b_cur, Y, 0);
    // conv_ltrb -> Y channels [128,640)
    conv1x1_wmma_kernel<<<dim3(HW / 64, CLTRB / 32, NBATCH), blk32, 0, stream>>>(
        feature, w_ltrb, b_ltrb, Y, CCUR);

    // InstanceNorm + ReLU: cur in place, ltrb -> channel-minor LT
    inorm_relu_kernel<<<dim3(NBATCH * CTOT, 1, 1), dim3(256, 1, 1), 0, stream>>>(Y, LT);

    // BorderAlign -> BA (torch permute+reshape layout)
    border_align_kernel<<<dim3(HW, 4, NBATCH), dim3(128, 1, 1), 0, stream>>>(
        LT, boxes, BA);

    // Final conv over concat + ReLU -> out
    border_out_wmma_kernel<<<dim3(HW / 64, MOUT / 32, NBATCH), blk32, 0, stream>>>(
        BA, Y, w_border, b_border, out);
}